// VAE_57655640981517
// MI455X (gfx1250) — compile-verified
//
#include <hip/hip_runtime.h>
#include <hip/hip_bf16.h>

// ---------------------------------------------------------------------------
// Sizes (match reference)
// ---------------------------------------------------------------------------
#define BB 64
#define TT 512
#define FF 1024
#define LL 128
#define NT (TT * FF)          // 524288, decoder N dimension
#define LDS_STRIDE 132        // padded row stride for zs tile in LDS

typedef __attribute__((ext_vector_type(2))) float v2f;
typedef __attribute__((ext_vector_type(8))) float v8f;

// ---------------------------------------------------------------------------
// threefry2x32 (20 rounds) — matches JAX's PRNG. Host+device.
// ---------------------------------------------------------------------------
__host__ __device__ inline unsigned rotl32(unsigned v, int n) {
  return (v << n) | (v >> (32 - n));
}

__host__ __device__ inline void threefry2x32(unsigned k0, unsigned k1,
                                             unsigned c0, unsigned c1,
                                             unsigned& o0, unsigned& o1) {
  unsigned ks2 = k0 ^ k1 ^ 0x1BD11BDAu;
  unsigned x0 = c0 + k0;
  unsigned x1 = c1 + k1;
#define TF_ROUND(r) { x0 += x1; x1 = rotl32(x1, r); x1 ^= x0; }
  TF_ROUND(13) TF_ROUND(15) TF_ROUND(26) TF_ROUND(6)
  x0 += k1;  x1 += ks2 + 1u;
  TF_ROUND(17) TF_ROUND(29) TF_ROUND(16) TF_ROUND(24)
  x0 += ks2; x1 += k0 + 2u;
  TF_ROUND(13) TF_ROUND(15) TF_ROUND(26) TF_ROUND(6)
  x0 += k0;  x1 += k1 + 3u;
  TF_ROUND(17) TF_ROUND(29) TF_ROUND(16) TF_ROUND(24)
  x0 += k1;  x1 += ks2 + 4u;
  TF_ROUND(13) TF_ROUND(15) TF_ROUND(26) TF_ROUND(6)
  x0 += ks2; x1 += k0 + 5u;
#undef TF_ROUND
  o0 = x0; o1 = x1;
}

// erfinv, single precision (Giles 2012) — what jax's erfinv lowers to in f32.
__device__ inline float erfinv_f(float u) {
  float w = -__logf((1.0f - u) * (1.0f + u));
  float p;
  if (w < 5.0f) {
    w = w - 2.5f;
    p = 2.81022636e-08f;
    p = fmaf(p, w, 3.43273939e-07f);
    p = fmaf(p, w, -3.5233877e-06f);
    p = fmaf(p, w, -4.39150654e-06f);
    p = fmaf(p, w, 0.00021858087f);
    p = fmaf(p, w, -0.00125372503f);
    p = fmaf(p, w, -0.00417768164f);
    p = fmaf(p, w, 0.246640727f);
    p = fmaf(p, w, 1.50140941f);
  } else {
    w = sqrtf(w) - 3.0f;
    p = -0.000200214257f;
    p = fmaf(p, w, 0.000100950558f);
    p = fmaf(p, w, 0.00134934322f);
    p = fmaf(p, w, -0.00367342844f);
    p = fmaf(p, w, 0.00573950773f);
    p = fmaf(p, w, -0.0076224613f);
    p = fmaf(p, w, 0.00943887047f);
    p = fmaf(p, w, 1.00167406f);
    p = fmaf(p, w, 2.83297682f);
  }
  return p * u;
}

// jax.nn.softplus(x) = logaddexp(x, 0)
__device__ inline float softplus_f(float x) {
  return fmaxf(x, 0.0f) + log1pf(expf(-fabsf(x)));
}

// ---------------------------------------------------------------------------
// Kernel 0: zero scratch (h accumulator [B*F] + count [B])
// ---------------------------------------------------------------------------
__global__ void vae_zero_kernel(float* __restrict__ p, int n) {
  int i = blockIdx.x * blockDim.x + threadIdx.x;
  if (i < n) p[i] = 0.0f;
}

// ---------------------------------------------------------------------------
// Kernel 1: masked mean-pool partial sums.
// grid (B, 8), block 256. Each block covers 64 timesteps, full F.
// ---------------------------------------------------------------------------
__global__ void vae_pool_kernel(const float* __restrict__ x,
                                float* __restrict__ h,     // [B*F] accum
                                float* __restrict__ cnt) { // [B] accum
  __shared__ float red[256];
  const int b = blockIdx.x;
  const int tid = threadIdx.x;
  const float* xb = x + (long)b * TT * FF;

  float a0 = 0.f, a1 = 0.f, a2 = 0.f, a3 = 0.f, c = 0.f;
  const int t0 = blockIdx.y * (TT / 8);
  for (int t = t0; t < t0 + TT / 8; ++t) {
    const float* row = xb + (long)t * FF;
    float v0 = row[tid];
    float v1 = row[tid + 256];
    float v2 = row[tid + 512];
    float v3 = row[tid + 768];
    red[tid] = (v0 + v1) + (v2 + v3);
    __syncthreads();
    for (int s = 128; s > 0; s >>= 1) {
      if (tid < s) red[tid] += red[tid + s];
      __syncthreads();
    }
    float rowsum = red[0];
    __syncthreads();
    if (rowsum != 0.0f) {
      a0 += v0; a1 += v1; a2 += v2; a3 += v3; c += 1.0f;
    }
  }
  atomicAdd(&h[b * FF + tid],       a0);
  atomicAdd(&h[b * FF + tid + 256], a1);
  atomicAdd(&h[b * FF + tid + 512], a2);
  atomicAdd(&h[b * FF + tid + 768], a3);
  if (tid == 0) atomicAdd(&cnt[b], c);
}

// ---------------------------------------------------------------------------
// Kernel 2: encoder heads + reparam sample.
// grid B, block 128 (thread = latent index l).
// ---------------------------------------------------------------------------
__global__ void vae_heads_kernel(const float* __restrict__ h,
                                 const float* __restrict__ cnt,
                                 const float* __restrict__ Wmu,
                                 const float* __restrict__ bmu,
                                 const float* __restrict__ Wstd,
                                 const float* __restrict__ bstd,
                                 const float* __restrict__ actionbias,
                                 const int*   __restrict__ label,
                                 float* __restrict__ out_mu,
                                 float* __restrict__ out_std,
                                 float* __restrict__ out_zs,
                                 unsigned fk0, unsigned fk1) {
  __shared__ float hs[FF];
  const int b = blockIdx.x;
  const int l = threadIdx.x;
  const float denom = fmaxf(cnt[b], 1.0f);
  for (int k = l; k < FF; k += LL) hs[k] = h[b * FF + k] / denom;
  __syncthreads();

  float accmu = bmu[l];
  float accsd = bstd[l];
  for (int k = 0; k < FF; ++k) {
    float hv = hs[k];
    accmu = fmaf(hv, Wmu[k * LL + l], accmu);
    accsd = fmaf(hv, Wstd[k * LL + l], accsd);
  }
  const float stdv = softplus_f(accsd);

  // eps = jax.random.normal(fold_in(key(0),1), (B,L)) element (b*L+l)
  const int idx = b * LL + l;
  const unsigned half = (idx < (BB * LL) / 2) ? 0u : 1u;
  const unsigned i = (unsigned)idx - half * ((BB * LL) / 2);
  unsigned o0, o1;
  threefry2x32(fk0, fk1, i, i + (BB * LL) / 2, o0, o1);
  const unsigned bits = half ? o1 : o0;
  union { unsigned u; float f; } cvt;
  cvt.u = (bits >> 9) | 0x3f800000u;             // [1,2)
  const float f01 = cvt.f - 1.0f;                // [0,1)
  const float lo = -0.99999994f;
  const float u = fmaxf(lo, fmaf(f01, 1.0f - lo, lo));
  const float eps = 1.41421356f * erfinv_f(u);

  const float zs = accmu + stdv * eps + actionbias[label[b] * LL + l];
  out_mu[idx]  = accmu;
  out_std[idx] = stdv;
  out_zs[idx]  = zs;
}

// ---------------------------------------------------------------------------
// Kernel 3: scale = softplus(log_scale)
// ---------------------------------------------------------------------------
__global__ void vae_scale_kernel(const float* __restrict__ ls,
                                 float* __restrict__ out) {
  int i = blockIdx.x * blockDim.x + threadIdx.x;
  if (i < FF) out[i] = softplus_f(ls[i]);
}

// ---------------------------------------------------------------------------
// Kernel 4: decoder GEMM  loc = zs[64,128] @ W_dec[128, 524288] + b_dec
// via V_WMMA_F32_16X16X4_F32. Block = 256 threads (8 waves), each wave owns
// 16 N-columns and computes all 4 M-tiles (reusing each B fragment 4x).
// ---------------------------------------------------------------------------
__global__ void vae_dec_kernel(const float* __restrict__ zs,
                               const float* __restrict__ Wdec,
                               const float* __restrict__ bdec,
                               float* __restrict__ loc) {
  __shared__ float As[BB * LDS_STRIDE];  // zs tile, padded rows (64x132 f32)
  const int tid = threadIdx.x;

  // Stage zs (64x128) into LDS with padded stride (kills bank conflicts on
  // the strided A-fragment reads).
  for (int i = tid; i < BB * LL; i += 256) {
    int r = i >> 7;           // /128
    int c = i & (LL - 1);
    As[r * LDS_STRIDE + c] = zs[i];
  }
  __syncthreads();

  const int wave = tid >> 5;
  const int lane = tid & 31;
  const int col  = lane & 15;
  const int half = lane >> 4;                    // 0: K pair {0,1}; 1: {2,3}
  const long n   = (long)blockIdx.x * 128 + wave * 16 + col;

  v8f acc0 = {}, acc1 = {}, acc2 = {}, acc3 = {};
  const float* Bp = Wdec + n;

#pragma unroll 4
  for (int k0 = 0; k0 < LL; k0 += 4) {
    const int kk = k0 + half * 2;
    v2f bfrag;
    bfrag.x = Bp[(long)kk * NT];
    bfrag.y = Bp[(long)(kk + 1) * NT];

#define MTILE(M, ACC)                                                        \
    {                                                                        \
      const v2f af = *(const v2f*)&As[((M) * 16 + col) * LDS_STRIDE + kk];   \
      ACC = __builtin_amdgcn_wmma_f32_16x16x4_f32(                           \
          false, af, false, bfrag, (short)0, ACC, false, false);             \
    }
    MTILE(0, acc0)
    MTILE(1, acc1)
    MTILE(2, acc2)
    MTILE(3, acc3)
#undef MTILE
  }

  const float bias = bdec[n];
#pragma unroll
  for (int v = 0; v < 8; ++v) {
    const int r = v + half * 8;                  // row within 16-row tile
    loc[(long)(r)      * NT + n] = acc0[v] + bias;
    loc[(long)(r + 16) * NT + n] = acc1[v] + bias;
    loc[(long)(r + 32) * NT + n] = acc2[v] + bias;
    loc[(long)(r + 48) * NT + n] = acc3[v] + bias;
  }
}

// ---------------------------------------------------------------------------
// Launch
// ---------------------------------------------------------------------------
extern "C" void kernel_launch(void* const* d_in, const int* in_sizes, int n_in,
                              void* d_out, int out_size, void* d_ws, size_t ws_size,
                              hipStream_t stream) {
  const float* x          = (const float*)d_in[0];
  const int*   label      = (const int*)  d_in[1];
  const float* actionbias = (const float*)d_in[2];
  const float* Wmu        = (const float*)d_in[3];
  const float* bmu        = (const float*)d_in[4];
  const float* Wstd       = (const float*)d_in[5];
  const float* bstd       = (const float*)d_in[6];
  const float* Wdec       = (const float*)d_in[7];
  const float* bdec       = (const float*)d_in[8];
  const float* logscale   = (const float*)d_in[9];

  float* out       = (float*)d_out;
  float* out_mu    = out;
  float* out_std   = out + BB * LL;
  float* out_zs    = out + 2 * BB * LL;
  float* out_loc   = out + 3 * BB * LL;
  float* out_scale = out + 3 * BB * LL + (long)BB * TT * FF;

  // Scratch: h accumulator [B*F] + cnt [B]. Prefer d_ws; fall back to the
  // tail of the loc output region (decoder overwrites it last).
  const int scratch_elems = BB * FF + BB;
  float* h;
  if (ws_size >= (size_t)scratch_elems * sizeof(float)) {
    h = (float*)d_ws;
  } else {
    h = out_loc + (long)BB * TT * FF - scratch_elems;
  }
  float* cnt = h + BB * FF;

  // Host-side fold_in(key(0), 1): threefry2x32((0,0), (0,1))
  unsigned fk0, fk1;
  threefry2x32(0u, 0u, 0u, 1u, fk0, fk1);

  vae_zero_kernel<<<(scratch_elems + 255) / 256, 256, 0, stream>>>(h, scratch_elems);
  vae_pool_kernel<<<dim3(BB, 8), 256, 0, stream>>>(x, h, cnt);
  vae_heads_kernel<<<BB, LL, 0, stream>>>(h, cnt, Wmu, bmu, Wstd, bstd,
                                          actionbias, label,
                                          out_mu, out_std, out_zs, fk0, fk1);
  vae_scale_kernel<<<(FF + 255) / 256, 256, 0, stream>>>(logscale, out_scale);
  vae_dec_kernel<<<NT / 128, 256, 0, stream>>>(out_zs, Wdec, bdec, out_loc);
}